// EnhancedGCN_49744311222746
// MI455X (gfx1250) — compile-verified
//
#include <hip/hip_runtime.h>
#include <hip/hip_bf16.h>
#include <math.h>

typedef __attribute__((ext_vector_type(2))) float v2f;
typedef __attribute__((ext_vector_type(8))) float v8f;

#define HID 128
#define EMB 32
#define IN_DIM 10
#define BN_EPS 1e-5f

// ---------------------------------------------------------------------------
// Degree / normalization
// ---------------------------------------------------------------------------
__global__ void k_deg_init(float* __restrict__ deg, int N) {
    int i = blockIdx.x * blockDim.x + threadIdx.x;
    if (i < N) deg[i] = 1.0f;   // self-loop weight
}

__global__ void k_deg_scatter(const int* __restrict__ eidx,
                              const float* __restrict__ ew,
                              float* __restrict__ deg, int E) {
    int e = blockIdx.x * blockDim.x + threadIdx.x;
    if (e < E) atomicAdd(&deg[eidx[E + e]], ew[e]);   // col = target
}

__global__ void k_deg_finish(float* __restrict__ deg, int N) {
    int i = blockIdx.x * blockDim.x + threadIdx.x;
    if (i < N) {
        float d = deg[i];
        deg[i] = (d > 0.0f) ? rsqrtf(d) : 0.0f;
    }
}

// ---------------------------------------------------------------------------
// combined[i, 0:32] = emb_table[i, :]
// combined[i, 32:64] = nan_to_num(x[i,:]) @ ft_W.T + ft_b
// ---------------------------------------------------------------------------
__global__ void k_combined(const float* __restrict__ x,
                           const float* __restrict__ emb,
                           const float* __restrict__ ftW,
                           const float* __restrict__ ftb,
                           float* __restrict__ comb, int N) {
    int idx = blockIdx.x * blockDim.x + threadIdx.x;
    if (idx >= N * 64) return;
    int i = idx >> 6;
    int j = idx & 63;
    float v;
    if (j < EMB) {
        v = emb[i * EMB + j];
    } else {
        int o = j - EMB;
        float s = ftb[o];
#pragma unroll
        for (int k = 0; k < IN_DIM; ++k) {
            float xv = x[i * IN_DIM + k];
            xv = isnan(xv) ? 0.0f : xv;
            s = fmaf(xv, ftW[o * IN_DIM + k], s);
        }
        v = s;
    }
    comb[idx] = v;
}

// ---------------------------------------------------------------------------
// FP32 WMMA GEMM: out[M x Ncols] = A[M x K] @ W[Ncols x K]^T  (+bias, relu)
// One wave per 16x64 output strip (4 column tiles); A fragment loaded once
// per K-step, reused across 4 WMMAs. Software-pipelined: step k+4's five
// fragment loads are issued BEFORE step k's WMMAs, so the WMMAs never wait
// on freshly issued loads (waits allow the next group to stay outstanding).
// EXEC all-ones (full tiles only); blockDim.x == 32.
//
// V_WMMA_F32_16X16X4_F32 operand layout (ISA 7.12.2):
//   A (16x4): lane L -> M = L&15, VGPR pair holds K = (L>>4)*2 + {0,1}
//   B (4x16): lane L -> N = L&15, VGPR pair holds K = (L>>4)*2 + {0,1}
//   C/D     : lane L -> N = L&15, VGPR v holds M = v + 8*(L>>4)
// B[k,n] = W[n,k], so both A and B fragments are contiguous float2 loads.
// ---------------------------------------------------------------------------
template <int K>
__global__ __launch_bounds__(32, 1)
void k_gemm_wmma_f32(const float* __restrict__ A, int lda,
                     const float* __restrict__ W, int ldw,
                     const float* __restrict__ bias,
                     float* __restrict__ out, int ldo,
                     int do_relu) {
    const int lane = threadIdx.x;        // 0..31
    const int half = lane >> 4;          // 0|1 -> K sub-offset
    const int l16  = lane & 15;

    const int rowBase = blockIdx.x * 16; // M tile
    const int colBase = blockIdx.y * 64; // 4 x 16 column tiles

    const float* aPtr = A + (size_t)(rowBase + l16) * lda + half * 2;
    const float* wPtr = W + (size_t)(colBase + l16) * ldw + half * 2;
    const size_t wS = (size_t)16 * ldw;

    // prologue: fragments for k = 0
    v2f a  = *(const v2f*)(aPtr);
    v2f b0 = *(const v2f*)(wPtr);
    v2f b1 = *(const v2f*)(wPtr + wS);
    v2f b2 = *(const v2f*)(wPtr + 2 * wS);
    v2f b3 = *(const v2f*)(wPtr + 3 * wS);

    v8f acc[4] = {};
#pragma unroll
    for (int k = 0; k < K; k += 4) {
        // prefetch next step's fragments (last step: re-read current, in-bounds)
        const int kn = (k + 4 < K) ? (k + 4) : k;
        v2f an  = *(const v2f*)(aPtr + kn);
        v2f b0n = *(const v2f*)(wPtr + kn);
        v2f b1n = *(const v2f*)(wPtr + wS + kn);
        v2f b2n = *(const v2f*)(wPtr + 2 * wS + kn);
        v2f b3n = *(const v2f*)(wPtr + 3 * wS + kn);

        acc[0] = __builtin_amdgcn_wmma_f32_16x16x4_f32(false, a, false, b0,
                                                       (short)0, acc[0], false, false);
        acc[1] = __builtin_amdgcn_wmma_f32_16x16x4_f32(false, a, false, b1,
                                                       (short)0, acc[1], false, false);
        acc[2] = __builtin_amdgcn_wmma_f32_16x16x4_f32(false, a, false, b2,
                                                       (short)0, acc[2], false, false);
        acc[3] = __builtin_amdgcn_wmma_f32_16x16x4_f32(false, a, false, b3,
                                                       (short)0, acc[3], false, false);
        a = an; b0 = b0n; b1 = b1n; b2 = b2n; b3 = b3n;
    }

#pragma unroll
    for (int t = 0; t < 4; ++t) {
        const int col = colBase + t * 16 + l16;
        const float bb = bias ? bias[col] : 0.0f;
#pragma unroll
        for (int v = 0; v < 8; ++v) {
            int m = rowBase + v + 8 * half;
            float val = acc[t][v] + bb;
            if (do_relu) val = fmaxf(val, 0.0f);
            out[(size_t)m * ldo + col] = val;
        }
    }
}

// ---------------------------------------------------------------------------
// agg[i,:] = xt[i,:] * dinv[i]^2      (self-loop term, w = 1)
// ---------------------------------------------------------------------------
__global__ void k_selfloop(const float* __restrict__ xt,
                           const float* __restrict__ dinv,
                           float* __restrict__ agg, int N) {
    int idx = blockIdx.x * blockDim.x + threadIdx.x;
    if (idx >= N * HID) return;
    int i = idx >> 7;
    float d = dinv[i];
    agg[idx] = xt[idx] * d * d;
}

// ---------------------------------------------------------------------------
// One wave per edge: agg[c,:] += xt[r,:] * (dinv[r] * w * dinv[c])
// Coalesced 128B reads + global_atomic_add_f32; stays in the 192MB L2.
// ---------------------------------------------------------------------------
__global__ void k_edge_scatter(const int* __restrict__ eidx,
                               const float* __restrict__ ew,
                               const float* __restrict__ dinv,
                               const float* __restrict__ xt,
                               float* __restrict__ agg, int E) {
    int e = blockIdx.x * (blockDim.x >> 5) + (threadIdx.x >> 5);
    if (e >= E) return;
    int lane = threadIdx.x & 31;
    int r = eidx[e];
    int c = eidx[E + e];
    float norm = dinv[r] * ew[e] * dinv[c];
    const float* src = xt + (size_t)r * HID;
    float* dst = agg + (size_t)c * HID;
#pragma unroll
    for (int s = 0; s < 4; ++s) {
        int j = s * 32 + lane;
        atomicAdd(dst + j, src[j] * norm);
    }
}

// ---------------------------------------------------------------------------
// h = BN(relu(agg + conv_b)) + h     (eval-mode BN, residual)
// ---------------------------------------------------------------------------
__global__ void k_post(const float* __restrict__ agg,
                       const float* __restrict__ cb,
                       const float* __restrict__ gamma,
                       const float* __restrict__ beta,
                       const float* __restrict__ mean,
                       const float* __restrict__ var,
                       float* __restrict__ h, int N) {
    int idx = blockIdx.x * blockDim.x + threadIdx.x;
    if (idx >= N * HID) return;
    int j = idx & 127;
    float v = fmaxf(agg[idx] + cb[j], 0.0f);
    v = (v - mean[j]) * rsqrtf(var[j] + BN_EPS) * gamma[j] + beta[j];
    h[idx] = v + h[idx];
}

// ---------------------------------------------------------------------------
// out[i] = clip(h[i,:] . lin_W + lin_b, -10, 10); one wave per node
// ---------------------------------------------------------------------------
__global__ void k_final(const float* __restrict__ h,
                        const float* __restrict__ linW,
                        const float* __restrict__ linb,
                        float* __restrict__ out, int N) {
    int node = blockIdx.x * (blockDim.x >> 5) + (threadIdx.x >> 5);
    int lane = threadIdx.x & 31;
    float s = 0.0f;
    if (node < N) {
#pragma unroll
        for (int st = 0; st < 4; ++st) {
            int j = st * 32 + lane;
            s = fmaf(h[(size_t)node * HID + j], linW[j], s);
        }
    }
#pragma unroll
    for (int off = 16; off > 0; off >>= 1)
        s += __shfl_down(s, off, 32);
    if (node < N && lane == 0)
        out[node] = fminf(fmaxf(s + linb[0], -10.0f), 10.0f);
}

// ---------------------------------------------------------------------------
extern "C" void kernel_launch(void* const* d_in, const int* in_sizes, int n_in,
                              void* d_out, int out_size, void* d_ws, size_t ws_size,
                              hipStream_t stream) {
    const float* x      = (const float*)d_in[0];
    const int*   eidx   = (const int*)  d_in[1];   // [2, E]: row then col
    const float* ew     = (const float*)d_in[2];
    const float* emb    = (const float*)d_in[3];
    const float* ftW    = (const float*)d_in[4];
    const float* ftb    = (const float*)d_in[5];
    const float* combW  = (const float*)d_in[6];
    const float* combB  = (const float*)d_in[7];
    const float* convW  = (const float*)d_in[8];   // [3,128,128]
    const float* convB  = (const float*)d_in[9];   // [3,128]
    const float* gamma  = (const float*)d_in[10];
    const float* beta   = (const float*)d_in[11];
    const float* mean   = (const float*)d_in[12];
    const float* var    = (const float*)d_in[13];
    const float* linW   = (const float*)d_in[14];
    const float* linb   = (const float*)d_in[15];
    float* out = (float*)d_out;

    const int N = in_sizes[3] / EMB;   // 50000
    const int E = in_sizes[2];         // 800000

    // workspace layout (floats)
    float* ws   = (float*)d_ws;
    float* dinv = ws;                       // N
    float* comb = dinv + N;                 // N*64
    float* h    = comb + (size_t)N * 64;    // N*128
    float* xt   = h    + (size_t)N * HID;   // N*128
    float* agg  = xt   + (size_t)N * HID;   // N*128

    const int tilesM = N / 16;              // N is a multiple of 16 (50000/16 = 3125)

    // 1) deg -> dinv
    k_deg_init<<<(N + 255) / 256, 256, 0, stream>>>(dinv, N);
    k_deg_scatter<<<(E + 255) / 256, 256, 0, stream>>>(eidx, ew, dinv, E);
    k_deg_finish<<<(N + 255) / 256, 256, 0, stream>>>(dinv, N);

    // 2) combined = [emb | x @ ft_W.T + ft_b]
    k_combined<<<((size_t)N * 64 + 255) / 256, 256, 0, stream>>>(x, emb, ftW, ftb, comb, N);

    // 3) h = relu(combined @ comb_W.T + comb_b)   [N,64]@[64,128]
    {
        dim3 grid(tilesM, HID / 64);
        k_gemm_wmma_f32<64><<<grid, 32, 0, stream>>>(comb, 64, combW, 64, combB,
                                                     h, HID, /*relu=*/1);
    }

    // 4) three GCN layers
    for (int l = 0; l < 3; ++l) {
        const float* Wl = convW + (size_t)l * HID * HID;
        const float* bl = convB + (size_t)l * HID;

        dim3 grid(tilesM, HID / 64);
        k_gemm_wmma_f32<HID><<<grid, 32, 0, stream>>>(h, HID, Wl, HID, nullptr,
                                                      xt, HID, /*relu=*/0);

        k_selfloop<<<((size_t)N * HID + 255) / 256, 256, 0, stream>>>(xt, dinv, agg, N);

        k_edge_scatter<<<(E + 7) / 8, 256, 0, stream>>>(eidx, ew, dinv, xt, agg, E);

        k_post<<<((size_t)N * HID + 255) / 256, 256, 0, stream>>>(agg, bl, gamma, beta,
                                                                  mean, var, h, N);
    }

    // 5) out = clip(h @ lin_W.T + lin_b)
    k_final<<<(N + 7) / 8, 256, 0, stream>>>(h, linW, linb, out, N);
}